// sample_and_group_18167711662781
// MI455X (gfx1250) — compile-verified
//
#include <hip/hip_runtime.h>

// CDNA5 / gfx1250, wave32. Uses V_WMMA_F32_16X16X4_F32 for both linear layers.
// Workspace requirement: ~264 MB (knn idx 2MB @0, bn/partials @2MB, y buffer 256MB @8MB).

typedef __attribute__((ext_vector_type(2))) float v2f;
typedef __attribute__((ext_vector_type(8))) float v8f;

#define B_    8
#define N_    8192
#define C_    64
#define S_    2048
#define K_    32
#define OUT_  128
#define ROWS_ (B_ * S_ * K_)   // 524288 rows into the MLP
#define MT_   (ROWS_ / 16)     // 32768 M-tiles of 16 rows
#define NB_   2048             // gemm grid size (each block does MT_/NB_ = 16 tiles)
#define EPS_  1e-5f
#define CNT_  ((float)ROWS_)

// workspace byte offsets
#define WSO_KNN  0u
#define WSO_BN1  (2u * 1024u * 1024u)
#define WSO_BN2  (WSO_BN1 + 1024u)
#define WSO_PART (WSO_BN1 + 4096u)          // NB_*256 floats = 2MB
#define WSO_Y    (8u * 1024u * 1024u)       // ROWS_*128 floats = 256MB

// ---------------------------------------------------------------- sampled_coor
__global__ void k_sample(const float* __restrict__ coor, const int* __restrict__ indx,
                         float* __restrict__ sc_out) {
  int t = blockIdx.x * blockDim.x + threadIdx.x;
  if (t >= B_ * S_ * 3) return;
  int c = t % 3;
  int s = (t / 3) % S_;
  int b = t / (3 * S_);
  sc_out[t] = coor[((size_t)b * N_ + indx[s]) * 3 + c];
}

// ---------------------------------------------------------------- brute-force KNN
// 128 queries per block (one batch per blockIdx.y); coor staged as float4 in LDS
// (one ds_load_b128 per candidate, broadcast address). Per-thread top-32 list in LDS.
__global__ void k_knn(const float* __restrict__ coor, const int* __restrict__ indx,
                      int* __restrict__ knn) {
  __shared__ float4 tpts[1024];
  __shared__ float  dl[32 * 128];
  __shared__ int    il[32 * 128];
  const int tid = threadIdx.x;                 // 0..127
  const int b   = blockIdx.y;
  const int s   = blockIdx.x * 128 + tid;
  const int si  = indx[s];
  const float qx = coor[((size_t)b * N_ + si) * 3 + 0];
  const float qy = coor[((size_t)b * N_ + si) * 3 + 1];
  const float qz = coor[((size_t)b * N_ + si) * 3 + 2];
  float worst = 3.4e38f;
  int   wp    = 0;
#pragma unroll
  for (int j = 0; j < 32; ++j) dl[j * 128 + tid] = 3.4e38f;
  for (int t = 0; t < N_ / 1024; ++t) {
    __syncthreads();
    for (int e = tid; e < 1024; e += 128) {
      const float* p = &coor[((size_t)b * N_ + t * 1024 + e) * 3];
      tpts[e] = make_float4(p[0], p[1], p[2], 0.f);
    }
    __syncthreads();
    for (int p = 0; p < 1024; ++p) {
      float4 c = tpts[p];
      float dx = c.x - qx, dy = c.y - qy, dz = c.z - qz;
      float d = dx * dx + dy * dy + dz * dz;
      if (d < worst) {
        dl[wp * 128 + tid] = d;
        il[wp * 128 + tid] = t * 1024 + p;
        float w = -1.f; int wj = 0;
#pragma unroll
        for (int j = 0; j < 32; ++j) {
          float v = dl[j * 128 + tid];
          if (v > w) { w = v; wj = j; }
        }
        worst = w; wp = wj;
      }
    }
  }
  const size_t base = ((size_t)(b * S_ + s)) * K_;
#pragma unroll
  for (int j = 0; j < 32; ++j) knn[base + j] = il[j * 128 + tid];
}

// ---------------------------------------------------------------- fused GEMM (+bias, +stats)
// LAYER 1: feat gather (x, knn) -> y1 = feat @ W1^T + b1, write y1, partial BN1 stats.
// LAYER 2: read y1, BN1+ReLU in staging -> y2 = h @ W2^T + b2, overwrite in place, BN2 stats.
// 256 threads = 8 waves; each wave owns 16 output channels; weights pinned in 64 VGPRs.
// Dual accumulators (even/odd K-chunks) for WMMA ILP.
template <int LAYER>
__global__ void k_gemm(const float* __restrict__ x, const int* __restrict__ indx,
                       const int* __restrict__ knn,
                       const float* __restrict__ W, const float* __restrict__ bias,
                       const float* __restrict__ bnprm,
                       float* __restrict__ ybuf, float* __restrict__ part) {
  __shared__ float4 feat4[512];               // 16 rows x 128 cols
  __shared__ float  stats[256];
  __shared__ int    rowS4[16], rowK4[16];     // float4 base offsets into x
  __shared__ float  sc[128], sh[128];
  const int tid  = threadIdx.x;
  const int lane = tid & 31;
  const int wv   = tid >> 5;
  const int o    = wv * 16 + (lane & 15);     // output channel for this lane
  const int koff = (lane >> 4) << 1;          // 0 (lanes 0-15) or 2 (lanes 16-31)

  // B fragments: B[k][n] = W[n][k]; lane half picks K pair -> contiguous float2 loads.
  v2f bf[32];
#pragma unroll
  for (int c = 0; c < 32; ++c)
    bf[c] = *(const v2f*)(W + (size_t)o * 128 + 4 * c + koff);
  const float bv = bias[o];

  stats[tid] = 0.f;
  if (LAYER == 2 && tid < 128) { sc[tid] = bnprm[tid]; sh[tid] = bnprm[128 + tid]; }
  __syncthreads();

  const float4* x4 = (const float4*)x;
  const float*  feat = (const float*)feat4;

  for (int mt = blockIdx.x; mt < MT_; mt += NB_) {
    if (LAYER == 1) {
      if (tid < 16) {
        int g = mt * 16 + tid;
        int b = g >> 16;                      // / (S_*K_)
        int s = (g >> 5) & (S_ - 1);
        rowS4[tid] = (b * N_ + indx[s]) * (C_ / 4);
        rowK4[tid] = (b * N_ + knn[g]) * (C_ / 4);
      }
      __syncthreads();
      {
        // One (row, col4) pair per thread: both feature halves from two float4 loads.
        int r = tid >> 4, c4 = tid & 15;
        float4 sv = x4[rowS4[r] + c4];
        float4 kv = x4[rowK4[r] + c4];
        feat4[r * 32 + c4] = sv;
        feat4[r * 32 + 16 + c4] =
            make_float4(kv.x - sv.x, kv.y - sv.y, kv.z - sv.z, kv.w - sv.w);
      }
    } else {
      const float4* src4 = (const float4*)(ybuf + (size_t)mt * 2048);
#pragma unroll
      for (int it = 0; it < 2; ++it) {
        int e4 = tid + 256 * it;              // 0..511
        float4 v = src4[e4];
        int ib = (e4 & 31) * 4;
        float4 h;
        h.x = fmaxf(0.f, v.x * sc[ib + 0] + sh[ib + 0]);
        h.y = fmaxf(0.f, v.y * sc[ib + 1] + sh[ib + 1]);
        h.z = fmaxf(0.f, v.z * sc[ib + 2] + sh[ib + 2]);
        h.w = fmaxf(0.f, v.w * sc[ib + 3] + sh[ib + 3]);
        feat4[e4] = h;
      }
    }
    __syncthreads();

    v8f acc0 = {}, acc1 = {};
    const float* arow = feat + (lane & 15) * 128 + koff;
#pragma unroll
    for (int c = 0; c < 32; c += 2) {
      v2f a0 = *(const v2f*)(arow + 4 * c);
      v2f a1 = *(const v2f*)(arow + 4 * c + 4);
      acc0 = __builtin_amdgcn_wmma_f32_16x16x4_f32(false, a0, false, bf[c],
                                                   (short)0, acc0, false, false);
      acc1 = __builtin_amdgcn_wmma_f32_16x16x4_f32(false, a1, false, bf[c + 1],
                                                   (short)0, acc1, false, false);
    }

    float s1 = 0.f, s2 = 0.f;
    v8f yv;
#pragma unroll
    for (int r = 0; r < 8; ++r) {
      float y = acc0[r] + acc1[r] + bv;
      yv[r] = y;
      s1 += y; s2 += y * y;
    }

    // D layout: vgpr r -> row r (lanes 0-15) / row r+8 (lanes 16-31), col o.
    const int rbase = mt * 16 + ((lane >> 4) << 3);
    float* dst = ybuf + (size_t)rbase * 128 + o;
#pragma unroll
    for (int r = 0; r < 8; ++r) dst[(size_t)r * 128] = yv[r];

    s1 += __shfl_xor(s1, 16, 32);
    s2 += __shfl_xor(s2, 16, 32);
    if (lane < 16) { stats[o] += s1; stats[128 + o] += s2; }
    __syncthreads();
  }
  __syncthreads();
  part[(size_t)blockIdx.x * 256 + tid] = stats[tid];   // deterministic partials
}

// ---------------------------------------------------------------- finalize BN params
__global__ void k_bnfin(const float* __restrict__ part, const float* __restrict__ g,
                        const float* __restrict__ be, float* __restrict__ bnprm) {
  int o = threadIdx.x;
  if (o >= 128) return;
  float sum = 0.f, sq = 0.f;
  for (int b = 0; b < NB_; ++b) {
    sum += part[(size_t)b * 256 + o];
    sq  += part[(size_t)b * 256 + 128 + o];
  }
  float mean = sum / CNT_;
  float var  = sq / CNT_ - mean * mean;
  float inv  = rsqrtf(var + EPS_);
  float scl  = g[o] * inv;
  bnprm[o]       = scl;
  bnprm[128 + o] = be[o] - mean * scl;
}

// ---------------------------------------------------------------- BN2 + ReLU + max over K
__global__ void k_maxpool(const float* __restrict__ ybuf, const float* __restrict__ bnprm,
                          float* __restrict__ out) {
  int g = blockIdx.x;       // b*S + s
  int o = threadIdx.x;      // 0..127
  float scl = bnprm[o], shf = bnprm[128 + o];
  const float* src = ybuf + (size_t)g * K_ * 128 + o;
  float m = -3.4e38f;
#pragma unroll 4
  for (int k = 0; k < K_; ++k) {
    float v = fmaxf(0.f, src[(size_t)k * 128] * scl + shf);
    m = fmaxf(m, v);
  }
  out[(size_t)g * 128 + o] = m;
}

// ---------------------------------------------------------------- launch
extern "C" void kernel_launch(void* const* d_in, const int* in_sizes, int n_in,
                              void* d_out, int out_size, void* d_ws, size_t ws_size,
                              hipStream_t stream) {
  const float* x    = (const float*)d_in[0];
  const float* coor = (const float*)d_in[1];
  const int*   indx = (const int*)d_in[2];
  const float* w1   = (const float*)d_in[3];
  const float* b1   = (const float*)d_in[4];
  const float* g1   = (const float*)d_in[5];
  const float* be1  = (const float*)d_in[6];
  const float* w2   = (const float*)d_in[7];
  const float* b2   = (const float*)d_in[8];
  const float* g2   = (const float*)d_in[9];
  const float* be2  = (const float*)d_in[10];
  float* out = (float*)d_out;
  char*  ws  = (char*)d_ws;

  int*   knn  = (int*)(ws + WSO_KNN);
  float* bn1  = (float*)(ws + WSO_BN1);
  float* bn2  = (float*)(ws + WSO_BN2);
  float* part = (float*)(ws + WSO_PART);
  float* ybuf = (float*)(ws + WSO_Y);
  float* sc_out = out + (size_t)B_ * S_ * OUT_;

  k_sample<<<dim3((B_ * S_ * 3 + 255) / 256), 256, 0, stream>>>(coor, indx, sc_out);
  k_knn<<<dim3(S_ / 128, B_), 128, 0, stream>>>(coor, indx, knn);
  k_gemm<1><<<NB_, 256, 0, stream>>>(x, indx, knn, w1, b1, nullptr, ybuf, part);
  k_bnfin<<<1, 128, 0, stream>>>(part, g1, be1, bn1);
  k_gemm<2><<<NB_, 256, 0, stream>>>(x, indx, knn, w2, b2, bn1, ybuf, part);
  k_bnfin<<<1, 128, 0, stream>>>(part, g2, be2, bn2);
  k_maxpool<<<B_ * S_, 128, 0, stream>>>(ybuf, bn2, out);
}